// Metric_31112743092670
// MI455X (gfx1250) — compile-verified
//
#include <hip/hip_runtime.h>

typedef float v2f __attribute__((ext_vector_type(2)));
typedef float v8f __attribute__((ext_vector_type(8)));

#define F_DIM 64
#define H_DIM 32
#define EMB9  9   // EMB * 3

// ---------------------------------------------------------------------------
// Kernel 1: per-vertex MLP -> metric (N x 9), fp32 WMMA (16x16x4_f32).
// One wave (32 lanes) per 16-vertex tile.
// A-frag layout (16x4 f32): v0 = K0 (lanes 0-15) / K2 (lanes 16-31), v1 = K1/K3.
// B-frag layout (4x16 f32): v0 = row K0/K2 striped over lanes, v1 = K1/K3.
// C/D layout (16x16 f32): VGPR v, lanes0-15 -> (M=v, N=lane); lanes16-31 -> (M=v+8).
// ---------------------------------------------------------------------------
__global__ __launch_bounds__(32) void mlp_metric_kernel(
    const float* __restrict__ features, const float* __restrict__ W1,
    const float* __restrict__ b1, const float* __restrict__ W2,
    const float* __restrict__ b2, float* __restrict__ metric, int n_verts)
{
  __shared__ float h_lds[16 * H_DIM];   // 2 KB: staging h (16x32) between GEMMs

  const int lane  = threadIdx.x;        // 0..31 (wave32)
  const int mrow  = lane & 15;          // 0..15
  const int khalf = (lane >> 4) * 2;    // 0 for lanes 0-15, 2 for lanes 16-31
  const int row0  = blockIdx.x * 16;

  // ---------------- GEMM1: h = relu(F[16x64] @ W1[64x32] + b1) --------------
  v8f acc0 = {};  // cols 0..15
  v8f acc1 = {};  // cols 16..31
  #pragma unroll
  for (int k = 0; k < F_DIM; k += 4) {
    int row = row0 + mrow;
    if (row >= n_verts) row = n_verts - 1;            // clamp (keeps EXEC full)
    const float* ap = features + (size_t)row * F_DIM + k + khalf;
    v2f a; a.x = ap[0]; a.y = ap[1];

    const float* bp = W1 + (size_t)(k + khalf) * H_DIM + mrow;
    v2f bb0; bb0.x = bp[0];      bb0.y = bp[H_DIM];
    v2f bb1; bb1.x = bp[16];     bb1.y = bp[H_DIM + 16];

    acc0 = __builtin_amdgcn_wmma_f32_16x16x4_f32(false, a, false, bb0,
                                                 (short)0, acc0, false, false);
    acc1 = __builtin_amdgcn_wmma_f32_16x16x4_f32(false, a, false, bb1,
                                                 (short)0, acc1, false, false);
  }

  // bias + ReLU, then scatter h (C-layout) into row-major LDS (A-layout source)
  const float bias0 = b1[mrow];
  const float bias1 = b1[16 + mrow];
  const int mbase = (lane < 16) ? 0 : 8;
  #pragma unroll
  for (int v = 0; v < 8; ++v) {
    float h0 = acc0[v] + bias0; h0 = h0 > 0.0f ? h0 : 0.0f;
    float h1 = acc1[v] + bias1; h1 = h1 > 0.0f ? h1 : 0.0f;
    const int m = mbase + v;
    h_lds[m * H_DIM + mrow]      = h0;
    h_lds[m * H_DIM + 16 + mrow] = h1;
  }
  __syncthreads();

  // ---------------- GEMM2: metric16 = h[16x32] @ W2[32x9] + b2 --------------
  // Pad W2 cols 9..15 with zeros; accumulate a single 16x16 tile.
  v8f acc2 = {};
  #pragma unroll
  for (int k = 0; k < H_DIM; k += 4) {
    const float* ap = &h_lds[mrow * H_DIM + k + khalf];
    v2f a; a.x = ap[0]; a.y = ap[1];

    v2f bb; bb.x = 0.0f; bb.y = 0.0f;
    if (mrow < EMB9) {
      bb.x = W2[(size_t)(k + khalf)     * EMB9 + mrow];
      bb.y = W2[(size_t)(k + khalf + 1) * EMB9 + mrow];
    }
    acc2 = __builtin_amdgcn_wmma_f32_16x16x4_f32(false, a, false, bb,
                                                 (short)0, acc2, false, false);
  }

  if (mrow < EMB9) {
    const float bias = b2[mrow];
    #pragma unroll
    for (int v = 0; v < 8; ++v) {
      const int row = row0 + mbase + v;
      if (row < n_verts)
        metric[(size_t)row * EMB9 + mrow] = acc2[v] + bias;
    }
  }
}

// ---------------------------------------------------------------------------
// Kernel 0: zero the degree accumulators (row_sum | col_sum contiguous).
// ---------------------------------------------------------------------------
__global__ void zero_kernel(float* __restrict__ p, int n)
{
  int i = blockIdx.x * blockDim.x + threadIdx.x;
  if (i < n) p[i] = 0.0f;
}

// ---------------------------------------------------------------------------
// Kernel 2: per-edge weight + degree scatter-adds + COO index emission.
// ---------------------------------------------------------------------------
__global__ void edge_w_kernel(
    const int* __restrict__ edges, const float* __restrict__ vertices,
    const float* __restrict__ metric, float* __restrict__ wbuf,
    float* __restrict__ row_sum, float* __restrict__ col_sum,
    int* __restrict__ out_idx, long long n_edges)
{
  long long e = (long long)blockIdx.x * blockDim.x + threadIdx.x;
  if (e >= n_edges) return;

  const int s = edges[e];
  const int d = edges[n_edges + e];

  const float tx = vertices[3 * (size_t)d + 0] - vertices[3 * (size_t)s + 0];
  const float ty = vertices[3 * (size_t)d + 1] - vertices[3 * (size_t)s + 1];
  const float tz = vertices[3 * (size_t)d + 2] - vertices[3 * (size_t)s + 2];

  const float* m = metric + 9 * (size_t)s;
  const float d0 = m[0] * tx + m[1] * ty + m[2] * tz;
  const float d1 = m[3] * tx + m[4] * ty + m[5] * tz;
  const float d2 = m[6] * tx + m[7] * ty + m[8] * tz;

  const float dist2 = d0 * d0 + d1 * d1 + d2 * d2;
  const float w = 1.0f / (1.0f + dist2);

  wbuf[e] = w;
  atomicAdd(&row_sum[s], w);
  atomicAdd(&col_sum[d], w);

  // indices (2, 2E): rows = [src, dst], cols = [dst, src]
  out_idx[e]               = s;
  out_idx[n_edges + e]     = d;
  out_idx[2 * n_edges + e] = d;
  out_idx[3 * n_edges + e] = s;
}

// ---------------------------------------------------------------------------
// Kernel 3: normalize -> values = 0.5 * [w/row_sum[src], w/col_sum[dst]].
// ---------------------------------------------------------------------------
__global__ void normalize_kernel(
    const int* __restrict__ edges, const float* __restrict__ wbuf,
    const float* __restrict__ row_sum, const float* __restrict__ col_sum,
    float* __restrict__ out_vals, long long n_edges)
{
  long long e = (long long)blockIdx.x * blockDim.x + threadIdx.x;
  if (e >= n_edges) return;

  const int s = edges[e];
  const int d = edges[n_edges + e];
  const float w = wbuf[e];

  out_vals[e]           = 0.5f * w / row_sum[s];
  out_vals[n_edges + e] = 0.5f * w / col_sum[d];
}

// ---------------------------------------------------------------------------
extern "C" void kernel_launch(void* const* d_in, const int* in_sizes, int n_in,
                              void* d_out, int out_size, void* d_ws, size_t ws_size,
                              hipStream_t stream)
{
  const float* features = (const float*)d_in[0];
  const float* vertices = (const float*)d_in[1];
  const float* W1       = (const float*)d_in[2];
  const float* b1       = (const float*)d_in[3];
  const float* W2       = (const float*)d_in[4];
  const float* b2       = (const float*)d_in[5];
  const int*   edges    = (const int*)d_in[6];

  const int       n_verts = in_sizes[1] / 3;
  const long long n_edges = (long long)in_sizes[6] / 2;

  // Workspace layout (floats): metric | w | row_sum | col_sum
  float* metric  = (float*)d_ws;
  float* wbuf    = metric + (size_t)n_verts * 9;
  float* row_sum = wbuf + n_edges;
  float* col_sum = row_sum + n_verts;   // contiguous with row_sum

  // Output layout: [indices: 4*E int32][values: 2*E f32]
  int*   out_idx  = (int*)d_out;
  float* out_vals = (float*)d_out + 4 * n_edges;

  const int nz = 2 * n_verts;
  zero_kernel<<<(nz + 255) / 256, 256, 0, stream>>>(row_sum, nz);

  const int tiles = (n_verts + 15) / 16;
  mlp_metric_kernel<<<tiles, 32, 0, stream>>>(features, W1, b1, W2, b2,
                                              metric, n_verts);

  const unsigned eblocks = (unsigned)((n_edges + 255) / 256);
  edge_w_kernel<<<eblocks, 256, 0, stream>>>(edges, vertices, metric, wbuf,
                                             row_sum, col_sum, out_idx, n_edges);
  normalize_kernel<<<eblocks, 256, 0, stream>>>(edges, wbuf, row_sum, col_sum,
                                                out_vals, n_edges);
}